// GNSUndir_70652212019847
// MI455X (gfx1250) — compile-verified
//
#include <hip/hip_runtime.h>
#include <math.h>

typedef float v2f __attribute__((ext_vector_type(2)));
typedef float v8f __attribute__((ext_vector_type(8)));
typedef unsigned int u32x4 __attribute__((ext_vector_type(4)));
typedef int i32x4 __attribute__((ext_vector_type(4)));
typedef int i32x8 __attribute__((ext_vector_type(8)));

#define BB 32
#define NN 1024
#define DD 10
#define HH 10
#define KK 30
#define BN (BB*NN)

__device__ __forceinline__ float lrelu(float x){ return x > 0.f ? x : 0.01f*x; }

// Select from LDS without exec divergence: load is always safe (clamped index),
// then mask via select -> v_cndmask, no s_and_saveexec around the WMMA stream.
__device__ __forceinline__ float ldsw(const float* W, int idx, bool valid){
  float x = W[valid ? idx : 0];
  return valid ? x : 0.f;
}

// ---- Tensor Data Mover: 1-row 2D tile, n contiguous fp32 global -> LDS ----
__device__ __forceinline__ void tdm_copy_f32(const float* __restrict__ src, float* dstLds, int n){
  unsigned long long ga = (unsigned long long)(uintptr_t)src;
  unsigned ldsOff = (unsigned)(uintptr_t)dstLds;   // low 32 bits of LDS aperture addr = byte offset
  u32x4 g0;
  g0.x = 1u;                                               // count=1, user descriptor
  g0.y = ldsOff;                                           // lds_addr (bytes)
  g0.z = (unsigned)ga;                                     // global_addr[31:0]
  g0.w = ((unsigned)(ga >> 32) & 0x01FFFFFFu) | 0x80000000u; // global_addr[56:32] | type=2
  i32x8 g1;
  g1[0] = 0x20000;                   // workgroup_mask=0, data_size=2 (4 bytes)
  g1[1] = (n & 0xFFFF) << 16;        // tensor_dim0[15:0] -> w1[31:16]
  g1[2] = ((n >> 16) & 0xFFFF) | (1 << 16); // tensor_dim0 hi | tensor_dim1=1
  g1[3] = (n & 0xFFFF) << 16;        // tile_dim0 = n
  g1[4] = 1;                         // tile_dim1 = 1, tile_dim2 = 0
  g1[5] = n;                         // tensor_dim0_stride = n
  g1[6] = 0;
  g1[7] = 0;
  i32x4 z4 = {0,0,0,0};
  i32x8 z8 = {0,0,0,0,0,0,0,0};
  __builtin_amdgcn_tensor_load_to_lds(g0, g1, z4, z4, z8, 0);
}

// ---------------- init kernels ----------------

__global__ void k_init_edges(const float* __restrict__ Yr, const float* __restrict__ Yi,
                             const int* __restrict__ eidx,
                             float* __restrict__ Gp, float* __restrict__ Bp, int E){
  int t = blockIdx.x*blockDim.x + threadIdx.x;
  if (t >= BB*E) return;
  int b = t / E, e = t % E;
  int ei = eidx[2*e], ej = eidx[2*e+1];
  size_t base = (size_t)b*NN*NN;
  Gp[t] = Yr[base + (size_t)ei*NN + ej];
  Bp[t] = Yi[base + (size_t)ei*NN + ej];
}

__global__ void k_init_nodes(const float* __restrict__ Yr, const float* __restrict__ Yi,
                             const float* __restrict__ V0,
                             float* __restrict__ dg, float* __restrict__ db,
                             float* __restrict__ v, float* __restrict__ th,
                             float* __restrict__ m, float* __restrict__ loss){
  int t = blockIdx.x*blockDim.x + threadIdx.x;
  if (t >= BN) return;
  int b = t / NN, n = t % NN;
  size_t dia = (size_t)b*NN*NN + (size_t)n*NN + n;
  dg[t] = Yr[dia]; db[t] = Yi[dia];
  v[t]  = V0[2*t]; th[t] = V0[2*t+1];
  for (int d = 0; d < DD; ++d) m[(size_t)t*DD + d] = 0.f;
  if (t < KK) loss[t] = 0.f;
}

// ---------------- per-iteration kernels ----------------

__global__ void k_phase1(const float* __restrict__ v, const float* __restrict__ th,
                         const float* __restrict__ dg, const float* __restrict__ db,
                         float* __restrict__ Vr, float* __restrict__ Vi,
                         float* __restrict__ Ir, float* __restrict__ Ii,
                         float* __restrict__ Mn){
  int t = blockIdx.x*blockDim.x + threadIdx.x;
  if (t >= BN) return;
  float vv = v[t], tt = th[t];
  float vr = vv*cosf(tt), vi = vv*sinf(tt);
  Vr[t] = vr; Vi[t] = vi;
  float g = dg[t], bb = db[t];
  Ir[t] = g*vr - bb*vi;
  Ii[t] = g*vi + bb*vr;
  for (int d = 0; d < DD; ++d) Mn[(size_t)t*DD + d] = 0.f;
}

__global__ void k_edge_cur(const int* __restrict__ eidx,
                           const float* __restrict__ Gp, const float* __restrict__ Bp,
                           const float* __restrict__ Vr, const float* __restrict__ Vi,
                           float* __restrict__ Ir, float* __restrict__ Ii, int E){
  int t = blockIdx.x*blockDim.x + threadIdx.x;
  if (t >= BB*E) return;
  int b = t / E, e = t % E;
  int i = eidx[2*e], j = eidx[2*e+1];
  float g = Gp[t], bb = Bp[t];
  int bi = b*NN + i, bj = b*NN + j;
  float vri = Vr[bi], vii = Vi[bi], vrj = Vr[bj], vij = Vi[bj];
  atomicAdd(&Ir[bi], g*vrj - bb*vij);
  atomicAdd(&Ii[bi], g*vij + bb*vrj);
  atomicAdd(&Ir[bj], g*vri - bb*vii);
  atomicAdd(&Ii[bj], g*vii + bb*vri);
}

__global__ void k_mismatch(const int* __restrict__ bt,
                           const float* __restrict__ Pset, const float* __restrict__ Qset,
                           const float* __restrict__ Vr, const float* __restrict__ Vi,
                           const float* __restrict__ Ir, const float* __restrict__ Ii,
                           float* __restrict__ dP, float* __restrict__ dQ,
                           float* __restrict__ loss, int k){
  __shared__ float red[256];
  int t = blockIdx.x*256 + threadIdx.x;
  float l = 0.f;
  if (t < BN){
    int bty = bt[t];
    float vr = Vr[t], vi = Vi[t], ir = Ir[t], ii = Ii[t];
    float P = vr*ir + vi*ii;
    float Q = vi*ir - vr*ii;
    float dp = (bty == 1) ? 0.f : (Pset[t] - P);
    float dq = (bty == 1 || bty == 2) ? 0.f : (Qset[t] - Q);
    dP[t] = dp; dQ[t] = dq;
    l = dp*dp + dq*dq;
  }
  red[threadIdx.x] = l;
  __syncthreads();
  for (int s = 128; s > 0; s >>= 1){
    if (threadIdx.x < s) red[threadIdx.x] += red[threadIdx.x + s];
    __syncthreads();
  }
  if (threadIdx.x == 0) atomicAdd(&loss[k], red[0]);
}

// edge MLP: 22 -> 10 (lrelu) -> 10, WMMA f32 16x16x4, one wave per 16 edges.
// Weights TDM-staged into LDS once per block.
__global__ void k_edge_mlp(const int* __restrict__ eidx,
                           const float* __restrict__ Gp, const float* __restrict__ Bp,
                           const float* __restrict__ m,
                           const float* __restrict__ w1, const float* __restrict__ b1,
                           const float* __restrict__ w2, const float* __restrict__ b2,
                           float* __restrict__ Mn, int E, int k){
  __shared__ float Xs[8][16][24];
  __shared__ float Hs[8][16][12];
  __shared__ float Wraw1[22*HH];   // 220
  __shared__ float Wraw2[HH*HH];   // 100
  int wv = threadIdx.x >> 5, lane = threadIdx.x & 31;
  int tile = blockIdx.x*8 + wv;
  int r = lane & 15, hf = lane >> 4;
  int BE = BB*E;

  // ---- TDM: stage this iteration's weights into LDS (wave 0 only) ----
  if (wv == 0){
    tdm_copy_f32(w1 + (size_t)k*22*HH, Wraw1, 22*HH);
    tdm_copy_f32(w2 + (size_t)k*HH*HH, Wraw2, HH*HH);
    __builtin_amdgcn_s_wait_tensorcnt(0);
  }

  // ---- stage 16x22 input tile (padded to 24) ----
  int ge = tile*16 + r;
  int gec = ge < BE ? ge : BE - 1;
  int b = gec / E, e = gec % E;
  int ei = eidx[2*e], ej = eidx[2*e+1];
  for (int c0 = 0; c0 < 12; ++c0){
    int c = hf*12 + c0;
    float val;
    if (c < 10)       val = m[((size_t)b*NN + ei)*DD + c];
    else if (c < 20)  val = m[((size_t)b*NN + ej)*DD + (c - 10)];
    else if (c == 20) val = Gp[gec];
    else if (c == 21) val = Bp[gec];
    else              val = 0.f;
    Xs[wv][r][c] = val;
  }
  __syncthreads();

  // ---- layer 1: K=24 (22 real), N=16 (10 real) ----
  v8f acc = {0.f,0.f,0.f,0.f,0.f,0.f,0.f,0.f};
  int n = r;
  bool ncol = (n < HH);
  #pragma unroll
  for (int ks = 0; ks < 6; ++ks){
    int kr = 4*ks + 2*hf;
    v2f a; a.x = Xs[wv][r][kr]; a.y = Xs[wv][r][kr+1];
    v2f bf;
    bf.x = ldsw(Wraw1, kr*HH + n,     ncol && (kr   < 22));
    bf.y = ldsw(Wraw1, (kr+1)*HH + n, ncol && (kr+1 < 22));
    acc = __builtin_amdgcn_wmma_f32_16x16x4_f32(false, a, false, bf, (short)0, acc, false, false);
  }

  // ---- bias + lrelu -> Hs (16x12, cols 10,11 = 0) ----
  const float* B1 = b1 + (size_t)k*HH;
  {
    int c = r;
    #pragma unroll
    for (int i = 0; i < 8; ++i){
      int row = i + 8*hf;
      if (c < HH)      Hs[wv][row][c] = lrelu(acc[i] + B1[c]);
      else if (c < 12) Hs[wv][row][c] = 0.f;
    }
  }
  __syncthreads();

  // ---- layer 2: K=12 (10 real), N=16 (10 real) ----
  v8f acc2 = {0.f,0.f,0.f,0.f,0.f,0.f,0.f,0.f};
  #pragma unroll
  for (int ks = 0; ks < 3; ++ks){
    int kr = 4*ks + 2*hf;
    v2f a; a.x = Hs[wv][r][kr]; a.y = Hs[wv][r][kr+1];
    v2f bf;
    bf.x = ldsw(Wraw2, kr*HH + n,     ncol && (kr   < HH));
    bf.y = ldsw(Wraw2, (kr+1)*HH + n, ncol && (kr+1 < HH));
    acc2 = __builtin_amdgcn_wmma_f32_16x16x4_f32(false, a, false, bf, (short)0, acc2, false, false);
  }

  // ---- scatter phi to both endpoints ----
  const float* B2 = b2 + (size_t)k*DD;
  int c = r;
  if (c < DD){
    float bias = B2[c];
    #pragma unroll
    for (int i = 0; i < 8; ++i){
      int row = i + 8*hf;
      int ge2 = tile*16 + row;
      if (ge2 < BE){
        int b2i = ge2 / E, e2 = ge2 % E;
        int ei2 = eidx[2*e2], ej2 = eidx[2*e2+1];
        float ph = acc2[i] + bias;
        atomicAdd(&Mn[((size_t)b2i*NN + ei2)*DD + c], ph);
        atomicAdd(&Mn[((size_t)b2i*NN + ej2)*DD + c], ph);
      }
    }
  }
}

__device__ __forceinline__ float bias1c(const float* TB1, const float* VB1, const float* MB1, int c){
  if (c < 10) return TB1[c];
  if (c < 20) return VB1[c - 10];
  if (c < 30) return MB1[c - 20];
  return 0.f;
}

// fused node MLPs: feats(24) -> combined hidden(30, padded 32) -> [dth, dv, dm0..9]
__global__ void k_node_mlp(const int* __restrict__ bt,
                           float* __restrict__ v, float* __restrict__ th, float* __restrict__ m,
                           const float* __restrict__ dP, const float* __restrict__ dQ,
                           const float* __restrict__ Mn,
                           const float* __restrict__ tw1, const float* __restrict__ tb1,
                           const float* __restrict__ tw2, const float* __restrict__ tb2,
                           const float* __restrict__ vw1, const float* __restrict__ vb1,
                           const float* __restrict__ vw2, const float* __restrict__ vb2,
                           const float* __restrict__ mw1, const float* __restrict__ mb1,
                           const float* __restrict__ mw2, const float* __restrict__ mb2,
                           int k){
  __shared__ float Xs[8][16][24];
  __shared__ float Hs[8][16][32];
  __shared__ float Wn1[3*24*HH];   // [region(th,v,m)][24][10]  = 720
  __shared__ float Wn2[120];       // [0,10) th_w2, [10,20) v_w2, [20,120) m_w2
  int wv = threadIdx.x >> 5, lane = threadIdx.x & 31;
  int tile = blockIdx.x*8 + wv;
  int r = lane & 15, hf = lane >> 4;
  int node = tile*16 + r;   // BN % 128 == 0, always valid

  // ---- TDM: stage weights (wave 0 only) ----
  if (wv == 0){
    tdm_copy_f32(tw1 + (size_t)k*24*HH, Wn1,           24*HH);
    tdm_copy_f32(vw1 + (size_t)k*24*HH, Wn1 + 24*HH,   24*HH);
    tdm_copy_f32(mw1 + (size_t)k*24*HH, Wn1 + 2*24*HH, 24*HH);
    tdm_copy_f32(tw2 + (size_t)k*HH,    Wn2,           HH);
    tdm_copy_f32(vw2 + (size_t)k*HH,    Wn2 + 10,      HH);
    tdm_copy_f32(mw2 + (size_t)k*HH*DD, Wn2 + 20,      HH*DD);
    __builtin_amdgcn_s_wait_tensorcnt(0);
  }

  // ---- stage feats ----
  for (int c0 = 0; c0 < 12; ++c0){
    int c = hf*12 + c0;
    float val;
    if (c == 0)      val = v[node];
    else if (c == 1) val = th[node];
    else if (c == 2) val = dP[node];
    else if (c == 3) val = dQ[node];
    else if (c < 14) val = m[(size_t)node*DD + (c - 4)];
    else             val = Mn[(size_t)node*DD + (c - 14)];
    Xs[wv][r][c] = val;
  }
  __syncthreads();

  // ---- layer1: K=24, N=32 (two 16-col tiles) ----
  int n0 = r, n1 = 16 + r;
  bool v0c = (n0 < 30), v1c = (n1 < 30);
  int base0 = (n0/10)*24*HH + (n0%10);
  int base1 = (n1/10)*24*HH + (n1%10);
  v8f acc0 = {0.f,0.f,0.f,0.f,0.f,0.f,0.f,0.f};
  v8f acc1 = {0.f,0.f,0.f,0.f,0.f,0.f,0.f,0.f};
  #pragma unroll
  for (int ks = 0; ks < 6; ++ks){
    int kr = 4*ks + 2*hf;
    v2f a; a.x = Xs[wv][r][kr]; a.y = Xs[wv][r][kr+1];
    v2f b0, b1;
    b0.x = ldsw(Wn1, base0 + kr*HH,     v0c);
    b0.y = ldsw(Wn1, base0 + (kr+1)*HH, v0c);
    b1.x = ldsw(Wn1, base1 + kr*HH,     v1c);
    b1.y = ldsw(Wn1, base1 + (kr+1)*HH, v1c);
    acc0 = __builtin_amdgcn_wmma_f32_16x16x4_f32(false, a, false, b0, (short)0, acc0, false, false);
    acc1 = __builtin_amdgcn_wmma_f32_16x16x4_f32(false, a, false, b1, (short)0, acc1, false, false);
  }

  // ---- bias + lrelu -> Hs (pad cols 30,31 to 0) ----
  const float* TB1 = tb1 + (size_t)k*HH;
  const float* VB1 = vb1 + (size_t)k*HH;
  const float* MB1 = mb1 + (size_t)k*HH;
  #pragma unroll
  for (int i = 0; i < 8; ++i){
    int row = i + 8*hf;
    Hs[wv][row][n0] = v0c ? lrelu(acc0[i] + bias1c(TB1, VB1, MB1, n0)) : 0.f;
    Hs[wv][row][n1] = v1c ? lrelu(acc1[i] + bias1c(TB1, VB1, MB1, n1)) : 0.f;
  }
  __syncthreads();

  // ---- layer2: K=32 (30 real), block-diagonal weights, N=16 (12 real) ----
  v8f acc2 = {0.f,0.f,0.f,0.f,0.f,0.f,0.f,0.f};
  #pragma unroll
  for (int ks = 0; ks < 8; ++ks){
    int kr0 = 4*ks + 2*hf;
    v2f a; a.x = Hs[wv][r][kr0]; a.y = Hs[wv][r][kr0+1];
    v2f bf;
    #pragma unroll
    for (int q = 0; q < 2; ++q){
      int kr = kr0 + q;
      bool vld; int idx;
      if (kr < 10)      { vld = (r == 0);            idx = kr; }
      else if (kr < 20) { vld = (r == 1);            idx = 10 + (kr - 10); }
      else if (kr < 30) { vld = (r >= 2 && r < 12);  idx = 20 + (kr - 20)*DD + (r - 2); }
      else              { vld = false;               idx = 0; }
      float w = ldsw(Wn2, idx, vld);
      if (q == 0) bf.x = w; else bf.y = w;
    }
    acc2 = __builtin_amdgcn_wmma_f32_16x16x4_f32(false, a, false, bf, (short)0, acc2, false, false);
  }

  // ---- epilogue: masked state updates ----
  int c = r;
  #pragma unroll
  for (int i = 0; i < 8; ++i){
    int row = i + 8*hf;
    int nd = tile*16 + row;
    int bty = bt[nd];
    float val = acc2[i];
    if (c == 0){
      if (bty != 1) th[nd] += val + tb2[k];
    } else if (c == 1){
      if (bty != 1 && bty != 2) v[nd] += val + vb2[k];
    } else if (c < 12){
      m[(size_t)nd*DD + (c - 2)] += tanhf(val + mb2[(size_t)k*DD + (c - 2)]);
    }
  }
}

__global__ void k_final(const float* __restrict__ v, const float* __restrict__ th,
                        const float* __restrict__ loss, float* __restrict__ out){
  int t = blockIdx.x*blockDim.x + threadIdx.x;
  if (t < BN){
    out[2*t]   = v[t];
    out[2*t+1] = th[t];
  }
  if (t == 0){
    float L = 0.f;
    for (int kk = 0; kk < KK; ++kk){
      float w = powf(0.96f, (float)(KK - 1 - kk));
      L += w * loss[kk] / (float)BN;
    }
    out[2*BN] = L;
  }
}

// ---------------- host launcher ----------------

extern "C" void kernel_launch(void* const* d_in, const int* in_sizes, int n_in,
                              void* d_out, int out_size, void* d_ws, size_t ws_size,
                              hipStream_t stream){
  const int*   bus  = (const int*)  d_in[0];
  const float* Yr   = (const float*)d_in[1];
  const float* Yi   = (const float*)d_in[2];
  const float* Pset = (const float*)d_in[3];
  const float* Qset = (const float*)d_in[4];
  const float* V0   = (const float*)d_in[5];
  const float* ew1  = (const float*)d_in[6];
  const float* eb1  = (const float*)d_in[7];
  const float* ew2  = (const float*)d_in[8];
  const float* eb2  = (const float*)d_in[9];
  const float* tw1  = (const float*)d_in[10];
  const float* tb1  = (const float*)d_in[11];
  const float* tw2  = (const float*)d_in[12];
  const float* tb2  = (const float*)d_in[13];
  const float* vw1  = (const float*)d_in[14];
  const float* vb1  = (const float*)d_in[15];
  const float* vw2  = (const float*)d_in[16];
  const float* vb2  = (const float*)d_in[17];
  const float* mw1  = (const float*)d_in[18];
  const float* mb1  = (const float*)d_in[19];
  const float* mw2  = (const float*)d_in[20];
  const float* mb2  = (const float*)d_in[21];
  const int*   eidx = (const int*)  d_in[22];

  int E  = in_sizes[22] / 2;
  int BE = BB * E;

  float* f  = (float*)d_ws;
  float* Gp = f; f += BE;
  float* Bp = f; f += BE;
  float* dg = f; f += BN;
  float* db = f; f += BN;
  float* v  = f; f += BN;
  float* th = f; f += BN;
  float* m  = f; f += (size_t)BN*DD;
  float* Vr = f; f += BN;
  float* Vi = f; f += BN;
  float* Ir = f; f += BN;
  float* Ii = f; f += BN;
  float* dP = f; f += BN;
  float* dQ = f; f += BN;
  float* Mn = f; f += (size_t)BN*DD;
  float* loss = f; f += KK;

  dim3 blk(256);
  k_init_edges<<<(BE + 255)/256, blk, 0, stream>>>(Yr, Yi, eidx, Gp, Bp, E);
  k_init_nodes<<<(BN + 255)/256, blk, 0, stream>>>(Yr, Yi, V0, dg, db, v, th, m, loss);

  int etiles  = (BE + 15) / 16;
  int eblocks = (etiles + 7) / 8;
  int nblocks = BN / 128;

  for (int k = 0; k < KK; ++k){
    k_phase1  <<<BN/256, blk, 0, stream>>>(v, th, dg, db, Vr, Vi, Ir, Ii, Mn);
    k_edge_cur<<<(BE + 255)/256, blk, 0, stream>>>(eidx, Gp, Bp, Vr, Vi, Ir, Ii, E);
    k_mismatch<<<BN/256, blk, 0, stream>>>(bus, Pset, Qset, Vr, Vi, Ir, Ii, dP, dQ, loss, k);
    k_edge_mlp<<<eblocks, blk, 0, stream>>>(eidx, Gp, Bp, m, ew1, eb1, ew2, eb2, Mn, E, k);
    k_node_mlp<<<nblocks, blk, 0, stream>>>(bus, v, th, m, dP, dQ, Mn,
                                            tw1, tb1, tw2, tb2,
                                            vw1, vb1, vw2, vb2,
                                            mw1, mb1, mw2, mb2, k);
  }
  k_final<<<BN/256, blk, 0, stream>>>(v, th, loss, (float*)d_out);
}